// LoRAPool_69638599737463
// MI455X (gfx1250) — compile-verified
//
#include <hip/hip_runtime.h>

typedef float v2f __attribute__((ext_vector_type(2)));
typedef float v8f __attribute__((ext_vector_type(8)));
typedef int   v4i_vs __attribute__((vector_size(16)));   // matches builtin param type

#define D_DIM   2048
#define E_DIM   8
#define R_DIM   16
#define ER      128     // E*R
#define TM      64      // tokens per workgroup
#define KC      64      // d-chunk for stage-1 (up-proj)
#define NC      64      // d-chunk for stage-2 (down-proj)
#define SCALE   2.0f    // alpha / rank = 32/16

#define HS 68           // h-tile LDS row stride (272B: 16B aligned, conflict-free)
#define WS 68           // W1-tile LDS row stride
#define YS 132          // y / W2-tile LDS row stride (528B: 16B aligned)

#if defined(__AMDGCN__) && \
    __has_builtin(__builtin_amdgcn_global_load_async_to_lds_b128) && \
    __has_builtin(__builtin_amdgcn_s_wait_asynccnt)
#define HAVE_ASYNC 1
#else
#define HAVE_ASYNC 0
#endif

#if HAVE_ASYNC
// direct global -> LDS copy, tracked by ASYNCcnt (gfx1250 async path)
#define CP16(gsrc, ldst)                                                     \
    __builtin_amdgcn_global_load_async_to_lds_b128(                          \
        (__attribute__((address_space(1))) v4i_vs*)(gsrc),                   \
        (__attribute__((address_space(3))) v4i_vs*)(ldst), 0, 0)
#else
#define CP16(gsrc, ldst)                                                     \
    do { *(float4*)(ldst) = *(const float4*)(gsrc); } while (0)
#endif

__device__ __forceinline__ void cp_wait_barrier() {
#if HAVE_ASYNC
    __builtin_amdgcn_s_wait_asynccnt(0);
#endif
    __syncthreads();
}

__global__ __launch_bounds__(256)
void lora_moe_kernel(const float* __restrict__ h,
                     const float* __restrict__ pL,
                     const float* __restrict__ A,   // [E][R][D] == [128][2048]
                     const float* __restrict__ Bw,  // [E][D][R]
                     float* __restrict__ out)
{
    __shared__ float sGate[TM * E_DIM];   //   2 KB
    __shared__ float sH[2][TM * HS];      //  34 KB  double-buffered h tile [64][64]
    __shared__ float sW[2][ER * WS];      //  68 KB  double-buffered W1 [128][64] / W2 [64][128]
    __shared__ float sY[TM * YS];         //  33 KB  gated xa [64][128]

    const int tid  = threadIdx.x;
    const int wave = tid >> 5;
    const int lane = tid & 31;
    const int l16  = lane & 15;
    const int hi   = lane >> 4;           // 0: lanes 0-15, 1: lanes 16-31
    const long tokBase = (long)blockIdx.x * TM;

    // ---------------- gates: top-2 of 8, pre-scaled ----------------
    if (tid < TM) {
        const float* p = pL + (tokBase + tid) * E_DIM;
        float v[E_DIM];
        #pragma unroll
        for (int e = 0; e < E_DIM; ++e) v[e] = p[e];
        int b1 = 0; float m1 = v[0];
        #pragma unroll
        for (int e = 1; e < E_DIM; ++e) if (v[e] > m1) { m1 = v[e]; b1 = e; }
        int b2 = -1; float m2 = -3.0e38f;
        #pragma unroll
        for (int e = 0; e < E_DIM; ++e) if (e != b1 && v[e] > m2) { m2 = v[e]; b2 = e; }
        #pragma unroll
        for (int e = 0; e < E_DIM; ++e) {
            float g = 0.0f;
            if (e == b1) g += m1;
            if (e == b2) g += m2;
            sGate[tid * E_DIM + e] = SCALE * g;
        }
    }

    // ---- async tile-copy issue helpers (per-thread, no barriers inside) ----
    auto issueH = [&](int kc, int p) {        // 64x64 floats, 4 x b128 per thread
        #pragma unroll
        for (int i = 0; i < 4; ++i) {
            int idx = tid + i * 256;
            int row = idx >> 4;
            int c4  = (idx & 15) << 2;
            CP16(h + (tokBase + row) * D_DIM + kc + c4, &sH[p][row * HS + c4]);
        }
    };
    auto issueW1 = [&](int kc, int p) {       // 128x64 floats, 8 x b128 per thread
        #pragma unroll
        for (int i = 0; i < 8; ++i) {
            int idx = tid + i * 256;
            int row = idx >> 4;
            int c4  = (idx & 15) << 2;
            CP16(A + (long)row * D_DIM + kc + c4, &sW[p][row * WS + c4]);
        }
    };
    auto issueW2 = [&](int nc, int p) {       // W2 tile [64 d][128 j], r contiguous
        #pragma unroll
        for (int i = 0; i < 8; ++i) {
            int q  = tid + i * 256;           // 0..2047
            int dl = q >> 5;                  // local d: 0..63
            int j4 = (q & 31) << 2;           // j = e*16 + r, r-aligned by 4
            int e  = j4 >> 4, r0 = j4 & 15;
            CP16(Bw + ((long)e * D_DIM + (nc + dl)) * R_DIM + r0,
                 &sW[p][dl * YS + j4]);
        }
    };

    // ---------------- stage 1: xa[64][128] = h_tile @ A^T ----------------
    // wave w owns rank-columns [16w, 16w+16) == expert w; 4 M-tiles each.
    issueH(0, 0);
    issueW1(0, 0);
    v8f acc[4] = {};
    for (int c = 0; c < D_DIM / KC; ++c) {
        const int p = c & 1;
        cp_wait_barrier();                    // chunk c landed in LDS, all waves see it
        if (c + 1 < D_DIM / KC) {             // overlap next copy with this compute
            issueH((c + 1) * KC, p ^ 1);
            issueW1((c + 1) * KC, p ^ 1);
        }
        const float* hB = &sH[p][0];
        const float* wB = &sW[p][0];
        #pragma unroll 4
        for (int k0 = 0; k0 < KC; k0 += 4) {
            const int kk = k0 + hi * 2;       // A/B VGPR pair: K = kk, kk+1
            v2f b = *(const v2f*)&wB[(wave * 16 + l16) * WS + kk];
            #pragma unroll
            for (int mt = 0; mt < 4; ++mt) {
                v2f a = *(const v2f*)&hB[(mt * 16 + l16) * HS + kk];
                acc[mt] = __builtin_amdgcn_wmma_f32_16x16x4_f32(
                    false, a, false, b, (short)0, acc[mt], false, false);
            }
        }
    }

    // ---------------- apply gates, stash y into LDS ----------------
    #pragma unroll
    for (int mt = 0; mt < 4; ++mt) {
        #pragma unroll
        for (int v = 0; v < 8; ++v) {
            int row = mt * 16 + v + hi * 8;           // C/D layout: M = v + 8*hi
            float g = sGate[row * E_DIM + wave];      // wave == expert for this N-tile
            sY[row * YS + wave * 16 + l16] = g * acc[mt][v];
        }
    }
    __syncthreads();   // sY visible; all stage-1 reads of sH/sW complete

    // ---------------- stage 2: out_tile = y @ W2, W2[j][d] = B[e][d][r] ----------------
    const int nt2 = wave & 3;             // d sub-tile within chunk
    const int mtb = (wave >> 2) << 1;     // token tiles {0,1} or {2,3}
    issueW2(0, 0);
    for (int c = 0; c < D_DIM / NC; ++c) {
        const int p  = c & 1;
        const int nc = c * NC;
        cp_wait_barrier();
        if (c + 1 < D_DIM / NC) issueW2(nc + NC, p ^ 1);

        const float* w2B = &sW[p][0];
        v8f o[2] = {};
        #pragma unroll 8
        for (int k0 = 0; k0 < ER; k0 += 4) {
            const int kk = k0 + hi * 2;
            v2f b = *(const v2f*)&w2B[(nt2 * 16 + l16) * YS + kk];
            #pragma unroll
            for (int i = 0; i < 2; ++i) {
                v2f a = *(const v2f*)&sY[((mtb + i) * 16 + l16) * YS + kk];
                o[i] = __builtin_amdgcn_wmma_f32_16x16x4_f32(
                    false, a, false, b, (short)0, o[i], false, false);
            }
        }
        #pragma unroll
        for (int i = 0; i < 2; ++i) {
            long rowBase = tokBase + (mtb + i) * 16 + hi * 8;
            #pragma unroll
            for (int v = 0; v < 8; ++v) {
                out[(rowBase + v) * D_DIM + nc + nt2 * 16 + l16] = o[i][v];
            }
        }
    }
}

extern "C" void kernel_launch(void* const* d_in, const int* in_sizes, int n_in,
                              void* d_out, int out_size, void* d_ws, size_t ws_size,
                              hipStream_t stream) {
    const float* h  = (const float*)d_in[0];
    const float* pL = (const float*)d_in[1];
    const float* A  = (const float*)d_in[2];
    const float* Bw = (const float*)d_in[3];
    float* out = (float*)d_out;

    const int nTok   = in_sizes[0] / D_DIM;   // Bt*S = 16384
    const int blocks = nTok / TM;             // 256 workgroups
    lora_moe_kernel<<<blocks, 256, 0, stream>>>(h, pL, A, Bw, out);
}